// KLLoss_74560632259496
// MI455X (gfx1250) — compile-verified
//
#include <hip/hip_runtime.h>

// KL( N(mu_p, diag(s_p^2)) || N(mu_q, diag(s_q^2)) ), summed over (N,D), meaned over (B,L).
// Bandwidth-bound streaming reduction (134 MB in, 4 B out @ 23.3 TB/s ~ 5.8us).
// Two-pass deterministic reduction; wave-level reduce uses V_WMMA_F32_16X16X4_F32:
//   A = ones(16x4), B = per-lane accumulators (2 per lane = 4x16 f32 B matrix)
//   => D[m][n] = sum_k B[k][n]; summing D row 0 over n gives the wave total.

typedef __attribute__((ext_vector_type(2))) float v2f;
typedef __attribute__((ext_vector_type(8))) float v8f;

#define NTHR 256
#define MAXBLK 1024

// Wave32 reduction via the matrix pipe. Every lane contributes acc0+acc1;
// returns the wave-wide total in all lanes. EXEC must be all 1s here.
__device__ __forceinline__ float wave_reduce_sum_wmma(float acc0, float acc1) {
    v2f a; a[0] = 1.0f; a[1] = 1.0f;      // A = 16x4 all-ones
    v2f b; b[0] = acc0; b[1] = acc1;      // B = 4x16 of wave partials (64 slots)
    v8f c = {};
    // 8 args: (neg_a, A, neg_b, B, c_mod, C, reuse_a, reuse_b)
    v8f d = __builtin_amdgcn_wmma_f32_16x16x4_f32(
        false, a, false, b, (short)0, c, false, false);
    // d[0]: lanes 0-15 hold D[0][n] = colsum(n); lanes 16-31 hold D[8][n] = colsum(n).
    // Butterfly over 16-lane halves: every lane ends with sum of all 16 colsums,
    // which equals the sum of all 64 B entries exactly once.
    float v = d[0];
    v += __shfl_xor(v, 8, 32);
    v += __shfl_xor(v, 4, 32);
    v += __shfl_xor(v, 2, 32);
    v += __shfl_xor(v, 1, 32);
    return v;
}

// 2x the per-element KL term (the 0.5 factor is folded into the final scale).
//   2*kl = r^2 + d^2 - 1 - 2*ln(r),  r = sp/sq,  d = (mq-mp)/sq
// v_rcp_f32 + v_log_f32 (log2; ln(r) = log2(r)*ln2) keep the VALU cost trivial.
__device__ __forceinline__ float kl_term2(float pm, float ps, float qm, float qs) {
    float inv = __builtin_amdgcn_rcpf(qs);
    float r   = ps * inv;
    float df  = (qm - pm) * inv;
    float l2  = __builtin_amdgcn_logf(r);            // log2(r)
    // r*r + df*df - 1 - 2*ln2*log2(r)
    return fmaf(r, r, fmaf(df, df, fmaf(-1.3862943611198906f, l2, -1.0f)));
}

__global__ void __launch_bounds__(NTHR)
kl_partial_kernel(const float* __restrict__ pmu, const float* __restrict__ psig,
                  const float* __restrict__ qmu, const float* __restrict__ qsig,
                  float* __restrict__ partials, int n) {
    const int nvec = n >> 2;
    const float4* pmu4  = (const float4*)pmu;
    const float4* psig4 = (const float4*)psig;
    const float4* qmu4  = (const float4*)qmu;
    const float4* qsig4 = (const float4*)qsig;

    float acc0 = 0.0f, acc1 = 0.0f;
    const int tid    = blockIdx.x * blockDim.x + threadIdx.x;
    const int stride = gridDim.x * blockDim.x;

    for (int i = tid; i < nvec; i += stride) {
        float4 pm = pmu4[i];     // 4x global_load_b128 per iteration
        float4 ps = psig4[i];
        float4 qm = qmu4[i];
        float4 qs = qsig4[i];
        acc0 += kl_term2(pm.x, ps.x, qm.x, qs.x);
        acc1 += kl_term2(pm.y, ps.y, qm.y, qs.y);
        acc0 += kl_term2(pm.z, ps.z, qm.z, qs.z);
        acc1 += kl_term2(pm.w, ps.w, qm.w, qs.w);
    }
    // Tail (n not multiple of 4): handled once, deterministically.
    if (blockIdx.x == 0 && threadIdx.x == 0) {
        for (int i = nvec << 2; i < n; ++i)
            acc0 += kl_term2(pmu[i], psig[i], qmu[i], qsig[i]);
    }

    // Full reconvergence here: EXEC all-ones for the WMMA.
    float wsum = wave_reduce_sum_wmma(acc0, acc1);

    __shared__ float wave_sums[NTHR / 32];
    const int lane = threadIdx.x & 31;
    const int wid  = threadIdx.x >> 5;
    if (lane == 0) wave_sums[wid] = wsum;
    __syncthreads();
    if (threadIdx.x == 0) {
        float s = 0.0f;
        #pragma unroll
        for (int w = 0; w < NTHR / 32; ++w) s += wave_sums[w];
        partials[blockIdx.x] = s;
    }
}

__global__ void kl_finalize_kernel(const float* __restrict__ partials, int nparts,
                                   float* __restrict__ out, float scale) {
    float acc = 0.0f;
    for (int i = threadIdx.x; i < nparts; i += 32) acc += partials[i];  // fixed order
    float total = wave_reduce_sum_wmma(acc, 0.0f);
    if (threadIdx.x == 0) out[0] = total * scale;   // scale = 0.5 / (B*L)
}

extern "C" void kernel_launch(void* const* d_in, const int* in_sizes, int n_in,
                              void* d_out, int out_size, void* d_ws, size_t ws_size,
                              hipStream_t stream) {
    const float* pmu  = (const float*)d_in[0];   // prior_mu
    const float* psig = (const float*)d_in[1];   // prior_sigma
    const float* qmu  = (const float*)d_in[2];   // post_mu
    const float* qsig = (const float*)d_in[3];   // post_sigma
    float* out = (float*)d_out;
    float* partials = (float*)d_ws;

    const int n = in_sizes[0];                   // B*L*N*D = 8,388,608
    // mean over (B,L): event dims are N=32, D=64 -> B*L = n / 2048
    const int BL = n / (32 * 64);
    const float scale = 0.5f / (float)(BL > 0 ? BL : 1);

    int nblk = MAXBLK;
    const int ws_cap = (int)(ws_size / sizeof(float));
    if (nblk > ws_cap) nblk = ws_cap > 0 ? ws_cap : 1;

    kl_partial_kernel<<<nblk, NTHR, 0, stream>>>(pmu, psig, qmu, qsig, partials, n);
    kl_finalize_kernel<<<1, 32, 0, stream>>>(partials, nblk, out, scale);
}